// MixL1SSIMLoss_35338990911645
// MI455X (gfx1250) — compile-verified
//
#include <hip/hip_runtime.h>
#include <math.h>

typedef float f2v __attribute__((ext_vector_type(2)));
typedef float v8f __attribute__((ext_vector_type(8)));
typedef unsigned int u32x4 __attribute__((ext_vector_type(4)));
typedef int i32x4 __attribute__((ext_vector_type(4)));
typedef int i32x8 __attribute__((ext_vector_type(8)));

#define XSTR 66            // 64-col tile + 2 pad (even -> 8B-aligned f2v rows)
#define SSIM_C1 1.0e-4f
#define SSIM_C2 9.0e-4f
#define MIX_ALPHA 0.985f

#if __has_builtin(__builtin_amdgcn_tensor_load_to_lds) && \
    __has_builtin(__builtin_amdgcn_s_wait_tensorcnt)
#define HAVE_TDM 1
#else
#define HAVE_TDM 0
#endif

#if HAVE_TDM
// D# group 0 (ISA 8.3): count=1 | lds_addr[63:32] | global_addr[120:64] | type=2[127:126]
__device__ __forceinline__ u32x4 tdm_g0(unsigned lds_addr, unsigned long long gaddr) {
    u32x4 g;
    g.x = 1u;                                            // count=1, user mode
    g.y = lds_addr;                                      // LDS byte address
    g.z = (unsigned)(gaddr & 0xFFFFFFFFull);             // global addr [95:64]
    g.w = (unsigned)((gaddr >> 32) & 0x01FFFFFFull)      // global addr [120:96]
          | (2u << 30);                                  // type = 2 ("image")
    return g;
}
// D# group 1 (ISA 8.4): 64x64 tile of 4B data, row stride 512, LDS pad 2 dw / 64 dw.
__device__ __forceinline__ i32x8 tdm_g1() {
    i32x8 g = {};
    g[0] = (2 << 16)        // data_size = 4 bytes
         | (1 << 20)        // pad_enable
         | (5 << 22)        // pad_interval: 2^(5+1) = 64 DWORDs
         | (1 << 25);       // pad_amount: 2 DWORDs  -> LDS row stride 66 = XSTR
    g[1] = (512 << 16);     // tensor_dim0[15:0] in [31:16]
    g[2] = (512 << 16);     // tensor_dim0[31:16]=0 | tensor_dim1[15:0]
    g[3] = (64 << 16);      // tensor_dim1[31:16]=0 | tile_dim0 = 64
    g[4] = 64;              // tile_dim1 = 64 | tile_dim2 = 0
    g[5] = 512;             // tensor_dim0_stride = 512 elements
    g[6] = 0;               // stride hi / dim1_stride lo
    g[7] = 0;               // dim1_stride hi
    return g;
}
#endif

// Each block: 128 threads = 4 waves, computes a 32x32 output region of one image.
// Each wave: one 16x16 output tile via Toeplitz-GEMM WMMA (f32 16x16x4).
__global__ __launch_bounds__(128)
void fused_mix_ssim_kernel(const float* __restrict__ x, const float* __restrict__ y,
                           float* __restrict__ partials)
{
    __shared__ __align__(16) float xt[64 * XSTR];
    __shared__ __align__(16) float yt[64 * XSTR];
    __shared__ __align__(16) float Tt[4][16 * 48];   // per-wave transposed horiz result
    __shared__ float gtab[5 * 33];
    __shared__ float red[128];

    const int tid   = threadIdx.x;
    const int n_img = blockIdx.x >> 8;        // 256 tiles per image
    const int t     = blockIdx.x & 255;
    const int r0    = (t >> 4) << 5;          // 32-row blocks
    const int c0    = (t & 15) << 5;          // 32-col blocks
    const float* xg = x + n_img * (512 * 512);
    const float* yg = y + n_img * (512 * 512);

    __builtin_prefetch(xg + r0 * 512 + c0, 0, 0);   // global_prefetch_b8
    __builtin_prefetch(yg + r0 * 512 + c0, 0, 0);

    const int lane = tid & 31;
    const int wv   = tid >> 5;        // wave id 0..3

    // ---- load 64x64 halo tiles into LDS ----
    const bool interior = (r0 != 0) && (r0 != 480) && (c0 != 0) && (c0 != 480);
#if HAVE_TDM
    if (interior) {
        // Whole halo is in-bounds: DMA both tiles with the Tensor Data Mover.
        if (wv == 0) {
            const float* gx = xg + (r0 - 16) * 512 + (c0 - 16);
            const float* gy = yg + (r0 - 16) * 512 + (c0 - 16);
            u32x4 g0x = tdm_g0((unsigned)(uintptr_t)&xt[0], (unsigned long long)(uintptr_t)gx);
            u32x4 g0y = tdm_g0((unsigned)(uintptr_t)&yt[0], (unsigned long long)(uintptr_t)gy);
            i32x8 g1  = tdm_g1();
            i32x4 z4  = {};
#if __clang_major__ >= 23
            i32x8 z8 = {};
            __builtin_amdgcn_tensor_load_to_lds(g0x, g1, z4, z4, z8, 0);
            __builtin_amdgcn_tensor_load_to_lds(g0y, g1, z4, z4, z8, 0);
#else
            __builtin_amdgcn_tensor_load_to_lds(g0x, g1, z4, z4, 0);
            __builtin_amdgcn_tensor_load_to_lds(g0y, g1, z4, z4, 0);
#endif
            __builtin_amdgcn_s_wait_tensorcnt((short)0);
        }
    } else
#endif
    {
        // Border tiles (or no TDM): predicated scalar loads, zero padding.
        for (int idx = tid; idx < 64 * 64; idx += 128) {
            int rr = idx >> 6, cc = idx & 63;
            int gr = r0 - 16 + rr, gc = c0 - 16 + cc;
            bool ok = (gr >= 0) & (gr < 512) & (gc >= 0) & (gc < 512);
            int go = gr * 512 + gc;
            xt[rr * XSTR + cc] = ok ? xg[go] : 0.0f;
            yt[rr * XSTR + cc] = ok ? yg[go] : 0.0f;
        }
    }

    // ---- build the 5 normalized 33-tap Gaussians in LDS ----
    if (tid < 33) {
        const float sig[5] = {0.5f, 1.0f, 2.0f, 4.0f, 8.0f};
        float c = (float)(tid - 16);
        for (int s2 = 0; s2 < 5; ++s2)
            gtab[s2 * 33 + tid] = __expf(-(c * c) / (2.0f * sig[s2] * sig[s2]));
    }
    __syncthreads();
    if (tid < 5) {
        float sum = 0.0f;
        for (int k = 0; k < 33; ++k) sum += gtab[tid * 33 + k];
        float inv = 1.0f / sum;
        for (int k = 0; k < 33; ++k) gtab[tid * 33 + k] *= inv;
    }
    __syncthreads();

    const int kh   = lane >> 4;       // half-wave selector
    const int lidx = lane & 15;
    const int lr   = (wv >> 1) << 4;  // 0 or 16 (tile row origin in region)
    const int lc   = (wv & 1) << 4;   // 0 or 16 (tile col origin in region)

    v8f csprod, ssim3, l1b;
    #pragma unroll
    for (int e = 0; e < 8; ++e) { csprod[e] = 1.0f; ssim3[e] = 0.0f; l1b[e] = 0.0f; }

    for (int s = 0; s < 5; ++s) {     // 5 distinct sigmas
        const float* gs = &gtab[s * 33];

        // Banded Toeplitz fragments: value(row j, lane n) = g[j-n].
        // Identical per-lane data serves as B (horizontal) and A (vertical).
        f2v gf[12];
        #pragma unroll
        for (int k = 0; k < 12; ++k) {
            int j  = 4 * k + 2 * kh;
            int d0 = j - lidx, d1 = d0 + 1;
            gf[k].x = (d0 >= 0 && d0 <= 32) ? gs[d0] : 0.0f;
            gf[k].y = (d1 >= 0 && d1 <= 32) ? gs[d1] : 0.0f;
        }

        // Separable 33-tap blur of derived field p at this wave's 16x16 tile.
        auto blur = [&](int p) -> v8f {
            float* tb = &Tt[wv][0];
            #pragma unroll
            for (int sub = 0; sub < 3; ++sub) {            // 48 rows of horiz result
                v8f acc = {};
                const int row = lr + sub * 16 + lidx;
                const float* xr = &xt[row * XSTR + lc];
                const float* yr = &yt[row * XSTR + lc];
                #pragma unroll
                for (int k = 0; k < 12; ++k) {
                    int o = 4 * k + 2 * kh;
                    f2v xv = *(const f2v*)(xr + o);
                    f2v a;
                    if (p == 0)      a = xv;
                    else if (p == 2) a = xv * xv;
                    else {
                        f2v yv = *(const f2v*)(yr + o);
                        if (p == 1)      a = yv;
                        else if (p == 3) a = yv * yv;
                        else if (p == 4) a = xv * yv;
                        else { f2v d = xv - yv;
                               a.x = __builtin_fabsf(d.x); a.y = __builtin_fabsf(d.y); }
                    }
                    acc = __builtin_amdgcn_wmma_f32_16x16x4_f32(
                            false, a, false, gf[k], (short)0, acc, false, false);
                }
                #pragma unroll
                for (int v = 0; v < 8; ++v)               // store transposed: Tt[n][i]
                    tb[lidx * 48 + sub * 16 + v + 8 * kh] = acc[v];
            }
            // vertical: Out = Gband(16x48) x T(48x16)
            v8f acc = {};
            #pragma unroll
            for (int k = 0; k < 12; ++k) {
                int o = 4 * k + 2 * kh;
                f2v bv = *(const f2v*)(tb + lidx * 48 + o);
                acc = __builtin_amdgcn_wmma_f32_16x16x4_f32(
                        false, gf[k], false, bv, (short)0, acc, false, false);
            }
            return acc;
        };

        v8f m_x  = blur(0);
        v8f m_y  = blur(1);
        v8f m_xx = blur(2);
        v8f m_yy = blur(3);
        v8f m_xy = blur(4);

        #pragma unroll
        for (int e = 0; e < 8; ++e) {
            float mx = m_x[e], my = m_y[e];
            float sx2 = m_xx[e] - mx * mx;
            float sy2 = m_yy[e] - my * my;
            float sxy = m_xy[e] - mx * my;
            float cs  = (2.0f * sxy + SSIM_C2) / (sx2 + sy2 + SSIM_C2);
            csprod[e] *= cs * cs * cs;                    // each sigma appears 3x
            if (s == 4) {
                float ss = (2.0f * mx * my + SSIM_C1) / (mx * mx + my * my + SSIM_C1);
                ssim3[e] = ss * ss * ss;
            }
        }
        if (s == 4) l1b = blur(5);                        // sigma=8 blur of |x-y|
    }

    float part = 0.0f;
    #pragma unroll
    for (int e = 0; e < 8; ++e)
        part += (1.0f - MIX_ALPHA) * (1.0f - ssim3[e] * csprod[e]) + MIX_ALPHA * l1b[e];

    red[tid] = part;
    __syncthreads();
    for (int st = 64; st > 0; st >>= 1) {
        if (tid < st) red[tid] += red[tid + st];
        __syncthreads();
    }
    if (tid == 0) partials[blockIdx.x] = red[0];
}

__global__ __launch_bounds__(256)
void final_reduce_kernel(const float* __restrict__ partials, float* __restrict__ out,
                         int nparts, float scale)
{
    __shared__ float red[256];
    int tid = threadIdx.x;
    float s = 0.0f;
    for (int i = tid; i < nparts; i += 256) s += partials[i];   // fixed order: deterministic
    red[tid] = s;
    __syncthreads();
    for (int st = 128; st > 0; st >>= 1) {
        if (tid < st) red[tid] += red[tid + st];
        __syncthreads();
    }
    if (tid == 0) out[0] = red[0] * scale;
}

extern "C" void kernel_launch(void* const* d_in, const int* in_sizes, int n_in,
                              void* d_out, int out_size, void* d_ws, size_t ws_size,
                              hipStream_t stream) {
    (void)in_sizes; (void)n_in; (void)out_size; (void)ws_size;
    const float* x = (const float*)d_in[0];
    const float* y = (const float*)d_in[1];
    float* partials = (float*)d_ws;
    float* out = (float*)d_out;

    const int nblocks = 8 * 16 * 16;   // 2048 tiles of 32x32 over [8,512,512]
    fused_mix_ssim_kernel<<<nblocks, 128, 0, stream>>>(x, y, partials);
    final_reduce_kernel<<<1, 256, 0, stream>>>(partials, out, nblocks,
                                               100.0f / (8.0f * 512.0f * 512.0f));
}